// Mann_ELT_30124900614610
// MI455X (gfx1250) — compile-verified
//
#include <hip/hip_runtime.h>

// ---------------------------------------------------------------------------
// Compile-time generation of the 2F1(1/3, -3/2, 4/3, w) power-series
// coefficients (Pfaff-transformed Mann series). c_n ~ n^{-3.5}; truncating at
// 40 terms leaves a relative tail of ~1e-5 at w=1 -- at/below the f32 noise of
// the reference's own 1000-term f32 scan.
// ---------------------------------------------------------------------------
#define NTERMS 40

struct Poly { float c[NTERMS]; };

constexpr Poly make_poly() {
    Poly p{};
    double term = 1.0;
    p.c[0] = 1.0f;
    for (int n = 0; n + 1 < NTERMS; ++n) {
        // term *= ((a+n)(bp+n)) / ((c+n)(n+1)),  a=1/3, bp=-3/2, c=4/3
        term *= ((1.0 / 3.0 + (double)n) * ((double)n - 1.5)) /
                ((4.0 / 3.0 + (double)n) * ((double)n + 1.0));
        p.c[n + 1] = (float)term;
    }
    return p;
}

constexpr Poly POLY = make_poly();

// ---------------------------------------------------------------------------
// tau = rsqrt(t2) * (1+t2)^(1/6) * rsqrt(S(w)),  w = 1/(1+t2),  t2 = L^2*|k|^2
// (algebraic simplification of  t^{-2/3} / sqrt((t2*w)^{1/3} * S(w)) )
// ---------------------------------------------------------------------------
__device__ __forceinline__ float mann_tau(float x, float y, float z,
                                          float L2, float g) {
    float t2 = fmaf(x, x, fmaf(y, y, z * z)) * L2;
    float u  = 1.0f + t2;
    float w  = __builtin_amdgcn_rcpf(u);          // v_rcp_f32
    float s  = POLY.c[NTERMS - 1];
#pragma unroll
    for (int i = NTERMS - 2; i >= 0; --i)
        s = fmaf(s, w, POLY.c[i]);                // Horner, constants folded
    // (1+t2)^{1/6} = exp2(log2(u)/6)
    float u16 = __builtin_amdgcn_exp2f(0.166666667f * __builtin_amdgcn_logf(u));
    float r   = __builtin_amdgcn_rsqf(t2) * __builtin_amdgcn_rsqf(s);
    float tau = g * u16 * r;
    return (t2 > 0.0f) ? tau : 0.0f;              // reference mask kL>0
}

// ---------------------------------------------------------------------------
// Main kernel: 256 threads/block (8 wave32), 4 elements/thread.
// Each wave async-copies its contiguous 1536B chunk of k into its private LDS
// region (3 x global_load_async_to_lds_b128, ASYNCcnt), waits, then each lane
// reads its own 48B (3 x ds_read_b128) -- AoS->per-thread gather via LDS with
// fully coalesced global traffic.
// ---------------------------------------------------------------------------
__global__ __launch_bounds__(256) void mann_main(const float* __restrict__ k,
                                                 const float* __restrict__ Lp,
                                                 const float* __restrict__ gp,
                                                 float* __restrict__ out) {
    __shared__ float4 sbuf[768];                  // 12 KB: mirror of block chunk

    const unsigned tid  = threadIdx.x;
    const unsigned lane = tid & 31u;
    const unsigned wave = tid >> 5;

    // uniform scalars (hoisted before the async wait)
    const float L  = Lp[0];
    const float L2 = L * L;
    const float g  = gp[0];

    // byte offset of this lane's first 16B chunk inside k (fits in u32: 100MB)
    unsigned goff = (unsigned)blockIdx.x * 12288u + wave * 1536u + lane * 16u;
    // matching LDS byte address (low 32 bits of flat shared addr = LDS offset)
    unsigned lds  = (unsigned)(size_t)(&sbuf[0]) + wave * 1536u + lane * 16u;

    asm volatile(
        "global_load_async_to_lds_b128 %0, %1, %2\n\t"
        "global_load_async_to_lds_b128 %0, %1, %2 offset:512\n\t"
        "global_load_async_to_lds_b128 %0, %1, %2 offset:1024\n\t"
        "s_wait_asynccnt 0"
        :
        : "v"(lds), "v"(goff), "s"(k)
        : "memory");

    // this thread's 12 floats = 4 (x,y,z) triples
    float4 p0 = sbuf[3u * tid + 0u];
    float4 p1 = sbuf[3u * tid + 1u];
    float4 p2 = sbuf[3u * tid + 2u];

    float4 o;
    o.x = mann_tau(p0.x, p0.y, p0.z, L2, g);
    o.y = mann_tau(p0.w, p1.x, p1.y, L2, g);
    o.z = mann_tau(p1.z, p1.w, p2.x, L2, g);
    o.w = mann_tau(p2.y, p2.z, p2.w, L2, g);

    reinterpret_cast<float4*>(out)[(unsigned)blockIdx.x * 256u + tid] = o;
}

// Scalar tail kernel (not used for the 256*256*128 shape, kept for generality)
__global__ void mann_tail(const float* __restrict__ k,
                          const float* __restrict__ Lp,
                          const float* __restrict__ gp,
                          float* __restrict__ out, int start, int n) {
    int i = start + (int)(blockIdx.x * blockDim.x + threadIdx.x);
    if (i < n) {
        float L = Lp[0];
        out[i] = mann_tau(k[3 * i], k[3 * i + 1], k[3 * i + 2], L * L, gp[0]);
    }
}

extern "C" void kernel_launch(void* const* d_in, const int* in_sizes, int n_in,
                              void* d_out, int out_size, void* d_ws, size_t ws_size,
                              hipStream_t stream) {
    (void)in_sizes; (void)n_in; (void)d_ws; (void)ws_size;
    const float* k = (const float*)d_in[0];
    const float* L = (const float*)d_in[1];
    const float* g = (const float*)d_in[2];
    float* out = (float*)d_out;

    const int n    = out_size;        // number of output elements
    const int nblk = n / 1024;        // 1024 elements per block
    if (nblk > 0)
        mann_main<<<nblk, 256, 0, stream>>>(k, L, g, out);
    const int done = nblk * 1024;
    const int rem  = n - done;
    if (rem > 0)
        mann_tail<<<(rem + 255) / 256, 256, 0, stream>>>(k, L, g, out, done, n);
}